// Encoder_7_7095285973677
// MI455X (gfx1250) — compile-verified
//
#include <hip/hip_runtime.h>
#include <hip/hip_bf16.h>

typedef __attribute__((ext_vector_type(2))) float v2f;
typedef __attribute__((ext_vector_type(8))) float v8f;

#define BATCH   32
#define TPAD    192
#define CH_CAT  768
#define NSEG    7
#define LIDX    (NSEG * 64)   // 7 segments * 2*MAX_LEN_SEG
#define XSTR    34            // padded LDS row stride (bank-conflict-free, 8B aligned)

// D = A(16x4,f32) * B(4x16,f32) + C  -- V_WMMA_F32_16X16X4_F32
#define WMMA_F32X4(Av, Bv, Cv) \
  __builtin_amdgcn_wmma_f32_16x16x4_f32(false, (Av), false, (Bv), (short)0, (Cv), false, false)

__device__ __forceinline__ float sigmoidf_(float x) { return 1.0f / (1.0f + __expf(-x)); }

// ---------------------------------------------------------------------------
// (B,337,192) channel-major -> (B,192,337) time-major
// ---------------------------------------------------------------------------
__global__ void transpose_xf0(const float* __restrict__ in, float* __restrict__ out) {
  int idx = blockIdx.x * blockDim.x + threadIdx.x;
  const int total = BATCH * 337 * TPAD;
  for (; idx < total; idx += gridDim.x * blockDim.x) {
    int b = idx / (337 * TPAD);
    int r = idx - b * 337 * TPAD;
    int c = r / TPAD;
    int t = r - c * TPAD;
    out[(size_t)(b * TPAD + t) * 337 + c] = in[idx];
  }
}

// ---------------------------------------------------------------------------
// Fused conv1d(k=5,pad=2) + bias + GroupNorm(16ch groups) + ReLU via f32 WMMA.
// One workgroup = (batch, group of 16 out channels). 4 waves, 3 t-tiles each
// (12 tiles x 16 = 192 timesteps). K staged in LDS in chunks of 32 channels.
// lds_x padded to stride 34 (conflict-free B reads); weights staged as
// [k][ci][co] so A reads are lane-consecutive (conflict-free).
// ---------------------------------------------------------------------------
__global__ void __launch_bounds__(128)
conv_gn_relu(const float* __restrict__ in, int in_stride, int in_choff, int Cin,
             const float* __restrict__ w, const float* __restrict__ bias,
             const float* __restrict__ gamma, const float* __restrict__ beta,
             float* __restrict__ out, int out_stride, int out_choff, int Cout_groups) {
  const int b   = blockIdx.x / Cout_groups;
  const int grp = blockIdx.x % Cout_groups;
  const int co0 = grp * 16;

  __shared__ float lds_x[196 * XSTR];   // t = -2..193, 32 channels (stride 34)
  __shared__ float lds_w[5 * 32 * 16];  // [k][ci][co]
  __shared__ float red1[128], red2[128];
  __shared__ float s_mean, s_rstd;

  const int tid  = threadIdx.x;
  const int lane = tid & 31;
  const int wave = tid >> 5;
  const int lm   = lane & 15;
  const int half = lane >> 4;

  v8f acc0 = {}; v8f acc1 = {}; v8f acc2 = {};

  const int nchunks = (Cin + 31) / 32;
  for (int chunk = 0; chunk < nchunks; ++chunk) {
    const int ci0 = chunk * 32;
    // stage input tile (zero-padded halo + tail channels), coalesced per row
    for (int e = tid; e < 196 * 32; e += 128) {
      int row = e >> 5, c = e & 31;
      int t = row - 2, ci = ci0 + c;
      float v = 0.0f;
      if (t >= 0 && t < TPAD && ci < Cin)
        v = in[(size_t)(b * TPAD + t) * in_stride + in_choff + ci];
      lds_x[row * XSTR + c] = v;
    }
    // stage weights: lds_w[(k*32 + ci)*16 + co] = w[co0+co][ci0+ci][k]
    for (int e = tid; e < 5 * 32 * 16; e += 128) {
      int co = e & 15;
      int tmp = e >> 4;
      int c = tmp & 31;
      int k = tmp >> 5;
      int ci = ci0 + c;
      float v = 0.0f;
      if (ci < Cin)
        v = w[((size_t)(co0 + co) * Cin + ci) * 5 + k];
      lds_w[e] = v;
    }
    __syncthreads();

    const int t0a = (wave * 3 + 0) * 16;
    const int t0b = (wave * 3 + 1) * 16;
    const int t0c = (wave * 3 + 2) * 16;
    for (int ci4 = 0; ci4 < 8; ++ci4) {
      const int cb = ci4 * 4 + half * 2;   // K-pair base for this lane-half
#pragma unroll
      for (int k = 0; k < 5; ++k) {
        v2f Av;
        Av.x = lds_w[(k * 32 + cb) * 16 + lm];
        Av.y = lds_w[(k * 32 + cb + 1) * 16 + lm];
        {
          int r = t0a + k + lm;            // lds row = t + 2, t = t0 + lm + k - 2
          v2f Bv; Bv.x = lds_x[r * XSTR + cb]; Bv.y = lds_x[r * XSTR + cb + 1];
          acc0 = WMMA_F32X4(Av, Bv, acc0);
        }
        {
          int r = t0b + k + lm;
          v2f Bv; Bv.x = lds_x[r * XSTR + cb]; Bv.y = lds_x[r * XSTR + cb + 1];
          acc1 = WMMA_F32X4(Av, Bv, acc1);
        }
        {
          int r = t0c + k + lm;
          v2f Bv; Bv.x = lds_x[r * XSTR + cb]; Bv.y = lds_x[r * XSTR + cb + 1];
          acc2 = WMMA_F32X4(Av, Bv, acc2);
        }
      }
    }
    __syncthreads();
  }

  // bias + GroupNorm statistics over the 16x192 group block
  float psum = 0.f, psq = 0.f;
#pragma unroll
  for (int j = 0; j < 8; ++j) {
    float bco = bias[co0 + j + half * 8];
    float v0 = acc0[j] + bco; psum += v0; psq += v0 * v0;
    float v1 = acc1[j] + bco; psum += v1; psq += v1 * v1;
    float v2 = acc2[j] + bco; psum += v2; psq += v2 * v2;
  }
  red1[tid] = psum; red2[tid] = psq;
  __syncthreads();
  for (int off = 64; off > 0; off >>= 1) {
    if (tid < off) { red1[tid] += red1[tid + off]; red2[tid] += red2[tid + off]; }
    __syncthreads();
  }
  if (tid == 0) {
    float inv  = 1.0f / (16.0f * (float)TPAD);
    float mean = red1[0] * inv;
    float var  = red2[0] * inv - mean * mean;
    s_mean = mean;
    s_rstd = rsqrtf(var + 1e-5f);
  }
  __syncthreads();
  const float mean = s_mean, rstd = s_rstd;
#pragma unroll
  for (int j = 0; j < 8; ++j) {
    int co = co0 + j + half * 8;
    float g = gamma[co], bt = beta[co], bco = bias[co];
    float vals[3] = { acc0[j] + bco, acc1[j] + bco, acc2[j] + bco };
#pragma unroll
    for (int tt = 0; tt < 3; ++tt) {
      float o = (vals[tt] - mean) * rstd * g + bt;
      o = fmaxf(o, 0.0f);
      int t = (wave * 3 + tt) * 16 + lm;
      out[(size_t)(b * TPAD + t) * out_stride + out_choff + co] = o;
    }
  }
}

// ---------------------------------------------------------------------------
// Random resampling (_interp). One workgroup per batch; per-segment cumsum and
// target positions computed sequentially in LDS, then parallel row scatter.
// in/out are (B,192,768) time-major. Rows without a source stay zero.
// ---------------------------------------------------------------------------
__global__ void __launch_bounds__(256)
interp_kernel(const float* __restrict__ in, const float* __restrict__ scales,
              const int* __restrict__ len_seg, float* __restrict__ out) {
  const int b = blockIdx.x;
  __shared__ int   s_tgt[LIDX];
  __shared__ int   s_ifl[LIDX];
  __shared__ float s_lam[LIDX];
  const int tid = threadIdx.x;

  if (tid == 0) {
    int segoff[NSEG]; int off = 0;
    for (int s = 0; s < NSEG; ++s) { segoff[s] = off; off += len_seg[b * NSEG + s]; }
    int cnt = 0;
    for (int e = 0; e < LIDX; ++e) {
      int seg = e >> 6, i = e & 63;
      float sc  = scales[b * NSEG + seg];
      float xs  = (float)i / sc;
      float fl  = floorf(xs);
      float lam = xs - fl;
      int   ls  = len_seg[b * NSEG + seg];
      float idx_org = fl + (float)segoff[seg];
      bool mask = (fl < (float)ls - 1.0f) && (idx_org < (float)(TPAD - 1));
      int tgt = -1;
      if (mask) { if (cnt < TPAD) tgt = cnt; cnt++; }
      int ifl = (int)idx_org;
      ifl = ifl < 0 ? 0 : (ifl > TPAD - 1 ? TPAD - 1 : ifl);
      s_tgt[e] = tgt; s_ifl[e] = ifl; s_lam[e] = lam;
    }
  }
  __syncthreads();

  float*       ob = out + (size_t)b * TPAD * CH_CAT;
  const float* ib = in  + (size_t)b * TPAD * CH_CAT;
  for (int p = tid; p < TPAD * CH_CAT; p += 256) ob[p] = 0.0f;
  __syncthreads();

  for (int e = 0; e < LIDX; ++e) {
    int tgt = s_tgt[e];
    if (tgt < 0) continue;
    int ifl = s_ifl[e];
    int icl = (ifl + 1 > TPAD - 1) ? TPAD - 1 : ifl + 1;
    float lam = s_lam[e];
    const float* r0 = ib + (size_t)ifl * CH_CAT;
    const float* r1 = ib + (size_t)icl * CH_CAT;
    float*       ro = ob + (size_t)tgt * CH_CAT;
    for (int c = tid; c < CH_CAT; c += 256)
      ro[c] = (1.0f - lam) * r0[c] + lam * r1[c];
  }
}

// ---------------------------------------------------------------------------
// C[M,N] = A[M,K] * W[N,K]^T via f32 WMMA. One workgroup per M-tile; one wave
// per N-tile. A chunk (16 x 64) staged coalesced into LDS (stride 68 keeps all
// 32 lanes on distinct banks). W read direct (tiny, L2-resident, reused by all
// M-tiles). K must be a multiple of 64.
// ---------------------------------------------------------------------------
#define ASTR 68
__global__ void gemm_wmma_nt(const float* __restrict__ A, int lda,
                             const float* __restrict__ W, int ldw,
                             float* __restrict__ C, int ldc, int K) {
  const int m0 = blockIdx.x * 16;
  const int n0 = (threadIdx.x >> 5) * 16;
  const int lane = threadIdx.x & 31;
  const int lm = lane & 15;
  const int half = lane >> 4;
  __shared__ float lds_a[16 * ASTR];
  v8f acc = {};
  const float* wrow = W + (size_t)(n0 + lm) * ldw + half * 2;
  for (int k0 = 0; k0 < K; k0 += 64) {
    for (int e = threadIdx.x; e < 16 * 64; e += blockDim.x) {
      int r = e >> 6, c = e & 63;
      const float* src = A + (size_t)(m0 + r) * lda + k0 + c;
      __builtin_prefetch(src + 64, 0, 1);
      lds_a[r * ASTR + c] = *src;
    }
    __syncthreads();
#pragma unroll
    for (int kk = 0; kk < 64; kk += 4) {
      v2f Av;
      Av.x = lds_a[lm * ASTR + kk + half * 2];
      Av.y = lds_a[lm * ASTR + kk + half * 2 + 1];
      v2f Bv;
      Bv.x = wrow[k0 + kk];
      Bv.y = wrow[k0 + kk + 1];
      acc = WMMA_F32X4(Av, Bv, acc);
    }
    __syncthreads();
  }
#pragma unroll
  for (int j = 0; j < 8; ++j)
    C[(size_t)(m0 + j + half * 8) * ldc + n0 + lm] = acc[j];
}

// ---------------------------------------------------------------------------
// LSTM recurrences (sequential over time, tiny hidden states).
// ---------------------------------------------------------------------------
__global__ void __launch_bounds__(32)
lstm_h8_proj(const float* __restrict__ xpf, const float* __restrict__ xpb,
             const float* __restrict__ whh_f, const float* __restrict__ whh_b,
             const float* __restrict__ bih_f, const float* __restrict__ bhh_f,
             const float* __restrict__ bih_b, const float* __restrict__ bhh_b,
             float* __restrict__ out /* (B,192,16) */) {
  const int b = blockIdx.x >> 1;
  const int dir = blockIdx.x & 1;
  const int tid = threadIdx.x;
  const float* xp  = dir ? xpb : xpf;
  const float* whh = dir ? whh_b : whh_f;
  const float bsum = dir ? (bih_b[tid] + bhh_b[tid]) : (bih_f[tid] + bhh_f[tid]);
  __shared__ float h_lds[8];
  __shared__ float g_lds[32];
  if (tid < 8) h_lds[tid] = 0.0f;
  float c = 0.0f;
  float wrow[8];
#pragma unroll
  for (int j = 0; j < 8; ++j) wrow[j] = whh[tid * 8 + j];
  __syncthreads();
  for (int s = 0; s < TPAD; ++s) {
    int t = dir ? (TPAD - 1 - s) : s;
    float v = xp[(size_t)(b * TPAD + t) * 32 + tid] + bsum;
#pragma unroll
    for (int j = 0; j < 8; ++j) v += wrow[j] * h_lds[j];
    g_lds[tid] = v;
    __syncthreads();
    if (tid < 8) {
      float ig = sigmoidf_(g_lds[tid]);
      float fg = sigmoidf_(g_lds[8 + tid]);
      float gg = tanhf(g_lds[16 + tid]);
      float og = sigmoidf_(g_lds[24 + tid]);
      c = fg * c + ig * gg;
      float h = og * tanhf(c);
      h_lds[tid] = h;
      out[(size_t)(b * TPAD + t) * 16 + dir * 8 + tid] = h;
    }
    __syncthreads();
  }
}

__global__ void __launch_bounds__(32)
lstm_h8_din16(const float* __restrict__ xin /* (B,192,16) */,
              const float* __restrict__ wih_f, const float* __restrict__ whh_f,
              const float* __restrict__ bih_f, const float* __restrict__ bhh_f,
              const float* __restrict__ wih_b, const float* __restrict__ whh_b,
              const float* __restrict__ bih_b, const float* __restrict__ bhh_b,
              float* __restrict__ out /* (B,192,16) */) {
  const int b = blockIdx.x >> 1;
  const int dir = blockIdx.x & 1;
  const int tid = threadIdx.x;
  const float* wih = dir ? wih_b : wih_f;
  const float* whh = dir ? whh_b : whh_f;
  const float bsum = dir ? (bih_b[tid] + bhh_b[tid]) : (bih_f[tid] + bhh_f[tid]);
  __shared__ float h_lds[8];
  __shared__ float x_lds[16];
  __shared__ float g_lds[32];
  if (tid < 8) h_lds[tid] = 0.0f;
  float c = 0.0f;
  float wi[16], wr[8];
#pragma unroll
  for (int k = 0; k < 16; ++k) wi[k] = wih[tid * 16 + k];
#pragma unroll
  for (int j = 0; j < 8; ++j) wr[j] = whh[tid * 8 + j];
  __syncthreads();
  for (int s = 0; s < TPAD; ++s) {
    int t = dir ? (TPAD - 1 - s) : s;
    if (tid < 16) x_lds[tid] = xin[(size_t)(b * TPAD + t) * 16 + tid];
    __syncthreads();
    float v = bsum;
#pragma unroll
    for (int k = 0; k < 16; ++k) v += wi[k] * x_lds[k];
#pragma unroll
    for (int j = 0; j < 8; ++j) v += wr[j] * h_lds[j];
    g_lds[tid] = v;
    __syncthreads();
    if (tid < 8) {
      float ig = sigmoidf_(g_lds[tid]);
      float fg = sigmoidf_(g_lds[8 + tid]);
      float gg = tanhf(g_lds[16 + tid]);
      float og = sigmoidf_(g_lds[24 + tid]);
      c = fg * c + ig * gg;
      float h = og * tanhf(c);
      h_lds[tid] = h;
      out[(size_t)(b * TPAD + t) * 16 + dir * 8 + tid] = h;
    }
    __syncthreads();
  }
}

__global__ void __launch_bounds__(128)
lstm_h32_proj(const float* __restrict__ xpf, const float* __restrict__ xpb,
              const float* __restrict__ whh_f, const float* __restrict__ whh_b,
              const float* __restrict__ bih_f, const float* __restrict__ bhh_f,
              const float* __restrict__ bih_b, const float* __restrict__ bhh_b,
              float* __restrict__ out /* (B,192,64) */) {
  const int b = blockIdx.x >> 1;
  const int dir = blockIdx.x & 1;
  const int tid = threadIdx.x;
  const float* xp  = dir ? xpb : xpf;
  const float* whh = dir ? whh_b : whh_f;
  const float bsum = dir ? (bih_b[tid] + bhh_b[tid]) : (bih_f[tid] + bhh_f[tid]);
  __shared__ float h_lds[32];
  __shared__ float g_lds[128];
  if (tid < 32) h_lds[tid] = 0.0f;
  float c = 0.0f;
  float wr[32];
#pragma unroll
  for (int j = 0; j < 32; ++j) wr[j] = whh[tid * 32 + j];
  __syncthreads();
  for (int s = 0; s < TPAD; ++s) {
    int t = dir ? (TPAD - 1 - s) : s;
    float v = xp[(size_t)(b * TPAD + t) * 128 + tid] + bsum;
#pragma unroll
    for (int j = 0; j < 32; ++j) v += wr[j] * h_lds[j];
    g_lds[tid] = v;
    __syncthreads();
    if (tid < 32) {
      float ig = sigmoidf_(g_lds[tid]);
      float fg = sigmoidf_(g_lds[32 + tid]);
      float gg = tanhf(g_lds[64 + tid]);
      float og = sigmoidf_(g_lds[96 + tid]);
      c = fg * c + ig * gg;
      float h = og * tanhf(c);
      h_lds[tid] = h;
      out[(size_t)(b * TPAD + t) * 64 + dir * 32 + tid] = h;
    }
    __syncthreads();
  }
}

// ---------------------------------------------------------------------------
// codes gather: d_out = codes_x (B,24,16) ++ codes_f0 (B,24,64)
// ---------------------------------------------------------------------------
__global__ void codes_kernel(const float* __restrict__ xo, const float* __restrict__ fo,
                             float* __restrict__ out) {
  const int NX = BATCH * 24 * 16;
  const int NF = BATCH * 24 * 64;
  int idx = blockIdx.x * blockDim.x + threadIdx.x;
  for (; idx < NX + NF; idx += gridDim.x * blockDim.x) {
    if (idx < NX) {
      int b = idx / (24 * 16);
      int r = (idx / 16) % 24;
      int c = idx % 16;
      int t = (c < 8) ? (7 + 8 * r) : (8 * r);
      out[idx] = xo[(size_t)(b * TPAD + t) * 16 + c];
    } else {
      int k = idx - NX;
      int b = k / (24 * 64);
      int r = (k / 64) % 24;
      int c = k % 64;
      int t = (c < 32) ? (7 + 8 * r) : (8 * r);
      out[idx] = fo[(size_t)(b * TPAD + t) * 64 + c];
    }
  }
}

// ---------------------------------------------------------------------------
extern "C" void kernel_launch(void* const* d_in, const int* in_sizes, int n_in,
                              void* d_out, int out_size, void* d_ws, size_t ws_size,
                              hipStream_t stream) {
  (void)in_sizes; (void)n_in; (void)out_size; (void)ws_size;
  const float* x_f0   = (const float*)d_in[0];
  const float* scales = (const float*)d_in[1];
  const int*   len_seg = (const int*)d_in[2];

  float* ws = (float*)d_ws;
  size_t o = 0;
  float* buf_tm0 = ws + o; o += (size_t)BATCH * TPAD * 337;
  float* bufA    = ws + o; o += (size_t)BATCH * TPAD * CH_CAT;
  float* bufB    = ws + o; o += (size_t)BATCH * TPAD * CH_CAT;
  float* xpf     = ws + o; o += (size_t)BATCH * TPAD * 32;
  float* xpb     = ws + o; o += (size_t)BATCH * TPAD * 32;
  float* fpf     = ws + o; o += (size_t)BATCH * TPAD * 128;
  float* fpb     = ws + o; o += (size_t)BATCH * TPAD * 128;
  float* xo0     = ws + o; o += (size_t)BATCH * TPAD * 16;
  float* xo      = ws + o; o += (size_t)BATCH * TPAD * 16;
  float* fo      = ws + o; o += (size_t)BATCH * TPAD * 64;

  transpose_xf0<<<512, 256, 0, stream>>>(x_f0, buf_tm0);

  for (int i = 0; i < 3; ++i) {
    const float* w1  = (const float*)d_in[3 + 4 * i + 0];
    const float* b1  = (const float*)d_in[3 + 4 * i + 1];
    const float* g1  = (const float*)d_in[3 + 4 * i + 2];
    const float* bt1 = (const float*)d_in[3 + 4 * i + 3];
    const float* w2  = (const float*)d_in[15 + 4 * i + 0];
    const float* b2  = (const float*)d_in[15 + 4 * i + 1];
    const float* g2  = (const float*)d_in[15 + 4 * i + 2];
    const float* bt2 = (const float*)d_in[15 + 4 * i + 3];

    const float* cin = (i == 0) ? buf_tm0 : bufB;
    int stride = (i == 0) ? 337 : CH_CAT;
    int foff   = (i == 0) ? 80 : 512;
    int xcin   = (i == 0) ? 80 : 512;
    int fcin   = (i == 0) ? 257 : 256;

    conv_gn_relu<<<BATCH * 32, 128, 0, stream>>>(cin, stride, 0, xcin,
                                                 w1, b1, g1, bt1, bufA, CH_CAT, 0, 32);
    conv_gn_relu<<<BATCH * 16, 128, 0, stream>>>(cin, stride, foff, fcin,
                                                 w2, b2, g2, bt2, bufA, CH_CAT, 512, 16);
    interp_kernel<<<BATCH, 256, 0, stream>>>(bufA, scales + i * BATCH * NSEG,
                                             len_seg + i * BATCH * NSEG, bufB);
  }

  // LSTM parameter pointers
  const float* wih1f = (const float*)d_in[27]; const float* whh1f = (const float*)d_in[28];
  const float* bih1f = (const float*)d_in[29]; const float* bhh1f = (const float*)d_in[30];
  const float* wih1b = (const float*)d_in[31]; const float* whh1b = (const float*)d_in[32];
  const float* bih1b = (const float*)d_in[33]; const float* bhh1b = (const float*)d_in[34];
  const float* wl1f_i = (const float*)d_in[35]; const float* wl1f_h = (const float*)d_in[36];
  const float* bl1f_i = (const float*)d_in[37]; const float* bl1f_h = (const float*)d_in[38];
  const float* wl1b_i = (const float*)d_in[39]; const float* wl1b_h = (const float*)d_in[40];
  const float* bl1b_i = (const float*)d_in[41]; const float* bl1b_h = (const float*)d_in[42];
  const float* wih2f = (const float*)d_in[43]; const float* whh2f = (const float*)d_in[44];
  const float* bih2f = (const float*)d_in[45]; const float* bhh2f = (const float*)d_in[46];
  const float* wih2b = (const float*)d_in[47]; const float* whh2b = (const float*)d_in[48];
  const float* bih2b = (const float*)d_in[49]; const float* bhh2b = (const float*)d_in[50];

  const int Mtiles = (BATCH * TPAD) / 16;  // 384
  // xx = bufB[:, :, 0:512], ff = bufB[:, :, 512:768]
  gemm_wmma_nt<<<Mtiles, 2 * 32, 0, stream>>>(bufB,       CH_CAT, wih1f, 512, xpf, 32,  512);
  gemm_wmma_nt<<<Mtiles, 2 * 32, 0, stream>>>(bufB,       CH_CAT, wih1b, 512, xpb, 32,  512);
  gemm_wmma_nt<<<Mtiles, 8 * 32, 0, stream>>>(bufB + 512, CH_CAT, wih2f, 256, fpf, 128, 256);
  gemm_wmma_nt<<<Mtiles, 8 * 32, 0, stream>>>(bufB + 512, CH_CAT, wih2b, 256, fpb, 128, 256);

  lstm_h8_proj<<<BATCH * 2, 32, 0, stream>>>(xpf, xpb, whh1f, whh1b,
                                             bih1f, bhh1f, bih1b, bhh1b, xo0);
  lstm_h8_din16<<<BATCH * 2, 32, 0, stream>>>(xo0, wl1f_i, wl1f_h, bl1f_i, bl1f_h,
                                              wl1b_i, wl1b_h, bl1b_i, bl1b_h, xo);
  lstm_h32_proj<<<BATCH * 2, 128, 0, stream>>>(fpf, fpb, whh2f, whh2b,
                                               bih2f, bhh2f, bih2b, bhh2b, fo);

  codes_kernel<<<240, 256, 0, stream>>>(xo, fo, (float*)d_out);
}